// MemoryBank_72164040508188
// MI455X (gfx1250) — compile-verified
//
#include <hip/hip_runtime.h>
#include <hip/hip_bf16.h>
#include <math.h>

typedef __attribute__((ext_vector_type(2))) float v2f;
typedef __attribute__((ext_vector_type(8))) float v8f;

namespace {
constexpr int CKd = 64;
constexpr int Qd  = 1620;   // 30*54 queries
constexpr int Od  = 3;
constexpr int CVd = 512;
constexpr int Md  = 32400;
constexpr int TK  = 20;
constexpr int NSLICE = 8;                        // split M across blocks
constexpr int QT  = 16;                          // WMMA N
constexpr int NQT = (Qd + QT - 1) / QT;          // 102 q tiles
constexpr int MT_PER_ITER = 128;                 // 8 waves * 16 m rows
constexpr int ITERS = 32;                        // 32*128 = 4096 m per slice
constexpr int SLICE_M = MT_PER_ITER * ITERS;     // 4096 ; 8*4096 >= 32400
constexpr int NNZ = Qd * TK;                     // 32400 sparse weights
constexpr int CVB = 4;                           // cv rows per readout block
constexpr float NEGINF = -3.0e38f;
}

// ---------------- kernel 0: a_sq[m] = sum_c mk[c,m]^2 ; zero histogram ----
__global__ void k_asq(const float* __restrict__ mk,
                      float* __restrict__ a_sq, int* __restrict__ count) {
  int m = blockIdx.x * blockDim.x + threadIdx.x;
  if (m >= Md) return;
  float s = 0.f;
  #pragma unroll 8
  for (int c = 0; c < CKd; ++c) { float v = mk[(size_t)c * Md + m]; s += v * v; }
  a_sq[m] = s;
  count[m] = 0;
}

// ---- kernel 1: fused affinity GEMM (f32 WMMA) + per-slice top-20 ---------
__global__ __launch_bounds__(256) void k_affinity_topk(
    const float* __restrict__ qk, const float* __restrict__ mk,
    const float* __restrict__ a_sq,
    float* __restrict__ part_v, int* __restrict__ part_i) {
  __shared__ float sv[256 * TK];
  __shared__ int   si[256 * TK];
  const int tid   = threadIdx.x;
  const int wave  = tid >> 5;
  const int lane  = tid & 31;
  const int half  = lane >> 4;
  const int l16   = lane & 15;
  const int qt    = blockIdx.x / NSLICE;
  const int slice = blockIdx.x % NSLICE;
  const int qbase = qt * QT;

  // Preload B (qk tile, 4x16 per WMMA step) -- loop invariant, lives in VGPRs.
  int qldr = qbase + l16; if (qldr >= Qd) qldr = Qd - 1;
  v2f breg[16];
  #pragma unroll
  for (int kk = 0; kk < 16; ++kk) {
    const int k0 = kk * 4 + half * 2;
    v2f b;
    b.x = qk[(size_t)k0 * Qd + qldr];
    b.y = qk[(size_t)(k0 + 1) * Qd + qldr];
    breg[kk] = b;
  }

  // Per-lane running top-20 (registers only: fully unrolled accesses).
  float tv[TK]; int ti[TK];
  #pragma unroll
  for (int i = 0; i < TK; ++i) { tv[i] = NEGINF; ti[i] = 0; }
  float runmin = NEGINF; int runpos = 0;

  const int mslice0 = slice * SLICE_M;
  for (int it = 0; it < ITERS; ++it) {
    const int mtile = mslice0 + it * MT_PER_ITER + wave * QT;
    if (mtile >= Md) break;                       // wave-uniform: EXEC stays full
    const int mlane = mtile + l16;
    const int mc = mlane < Md ? mlane : (Md - 1);

    v8f c = {};
    #pragma unroll
    for (int kk = 0; kk < 16; ++kk) {
      const int k0 = kk * 4 + half * 2;
      v2f a;
      a.x = mk[(size_t)k0 * Md + mc];
      a.y = mk[(size_t)(k0 + 1) * Md + mc];
      // D = A(16x4) * B(4x16) + C, f32 throughout
      c = __builtin_amdgcn_wmma_f32_16x16x4_f32(false, a, false, breg[kk],
                                                (short)0, c, false, false);
    }

    // lane holds 8 values of one q column: rows m = mtile + half*8 + r
    const int mrow0 = mtile + half * 8;
    #pragma unroll
    for (int r = 0; r < 8; ++r) {
      const int mr  = mrow0 + r;
      const int mrc = mr < Md ? mr : (Md - 1);
      float val = 0.25f * c[r] - 0.125f * a_sq[mrc];  // (2ab - asq)/sqrt(64)
      if (mr >= Md) val = NEGINF;
      if (val > runmin) {
        #pragma unroll
        for (int i = 0; i < TK; ++i) if (i == runpos) { tv[i] = val; ti[i] = mr; }
        runmin = tv[0]; runpos = 0;
        #pragma unroll
        for (int i = 1; i < TK; ++i) if (tv[i] < runmin) { runmin = tv[i]; runpos = i; }
      }
    }
  }

  // Merge: each q column is owned by 16 lanes (2 per wave x 8 waves).
  #pragma unroll
  for (int i = 0; i < TK; ++i) { sv[tid * TK + i] = tv[i]; si[tid * TK + i] = ti[i]; }
  __syncthreads();
  if (tid < QT) {
    const int qg = qbase + tid;
    if (qg < Qd) {
      const size_t outb = ((size_t)qg * NSLICE + slice) * TK;
      for (int k = 0; k < TK; ++k) {
        float best = NEGINF; int bpos = 0;
        for (int ow = 0; ow < 16; ++ow) {
          const int t = ((ow >> 1) << 5) + ((ow & 1) << 4) + tid;
          #pragma unroll
          for (int s2 = 0; s2 < TK; ++s2) {
            const float v = sv[t * TK + s2];
            if (v > best) { best = v; bpos = t * TK + s2; }
          }
        }
        part_v[outb + k] = best;
        part_i[outb + k] = si[bpos];
        sv[bpos] = NEGINF;
      }
    }
  }
}

// ---- kernel 2: merge 8 slices -> exact top-20, softmax, histogram --------
__global__ __launch_bounds__(32) void k_finalize(
    const float* __restrict__ part_v, const int* __restrict__ part_i,
    int* __restrict__ topk_idx, float* __restrict__ topk_w,
    int* __restrict__ count) {
  constexpr int NC = NSLICE * TK;                 // 160 candidates per q
  __shared__ float fv[32 * NC];
  __shared__ int   fi[32 * NC];
  __shared__ float selv[32 * TK];
  __shared__ int   seli[32 * TK];
  const int tid = threadIdx.x;
  const int q = blockIdx.x * 32 + tid;
  if (q >= Qd) return;                            // no barriers below
  float* mv = &fv[tid * NC];
  int*   mi = &fi[tid * NC];
  const size_t base = (size_t)q * NC;
  for (int i = 0; i < NC; ++i) { mv[i] = part_v[base + i]; mi[i] = part_i[base + i]; }
  for (int k = 0; k < TK; ++k) {
    float best = NEGINF; int bp = 0;
    for (int j = 0; j < NC; ++j) { const float v = mv[j]; if (v > best) { best = v; bp = j; } }
    selv[tid * TK + k] = best; seli[tid * TK + k] = mi[bp]; mv[bp] = NEGINF;
  }
  const float mx = selv[tid * TK + 0];            // first pick is the max
  float ssum = 0.f;
  for (int k = 0; k < TK; ++k) {
    const float e = __expf(selv[tid * TK + k] - mx);
    selv[tid * TK + k] = e; ssum += e;
  }
  const float inv = 1.f / ssum;
  for (int k = 0; k < TK; ++k) {
    const int m = seli[tid * TK + k];
    topk_idx[q * TK + k] = m;
    topk_w[q * TK + k] = selv[tid * TK + k] * inv;
    atomicAdd(&count[m], 1);
  }
}

// ---- kernel 3: exclusive scan of histogram -> scatter cursors ------------
__global__ __launch_bounds__(1024) void k_scan(int* __restrict__ count) {
  __shared__ int lds[1024];
  const int t = threadIdx.x;
  int loc[32];
  int sum = 0;
  #pragma unroll
  for (int j = 0; j < 32; ++j) {
    const int idx = t * 32 + j;
    const int v = (idx < Md) ? count[idx] : 0;
    loc[j] = v; sum += v;
  }
  lds[t] = sum; __syncthreads();
  for (int off = 1; off < 1024; off <<= 1) {
    const int v = (t >= off) ? lds[t - off] : 0;
    __syncthreads();
    lds[t] += v;
    __syncthreads();
  }
  int run = (t > 0) ? lds[t - 1] : 0;
  #pragma unroll
  for (int j = 0; j < 32; ++j) {
    const int idx = t * 32 + j;
    if (idx < Md) count[idx] = run;
    run += loc[j];
  }
}

// ---- kernel 4: counting-sort scatter of nonzeros by m --------------------
__global__ void k_scatter(const int* __restrict__ topk_idx,
                          const float* __restrict__ topk_w,
                          int* __restrict__ count,
                          int* __restrict__ e_mq, float* __restrict__ e_w) {
  const int i = blockIdx.x * blockDim.x + threadIdx.x;
  if (i >= NNZ) return;
  const int q = i / TK;
  const int m = topk_idx[i];
  const float w = topk_w[i];
  const int pos = atomicAdd(&count[m], 1);
  e_mq[pos] = m | (q << 15);                      // m < 2^15, q < 2^11
  e_w[pos] = w;
}

// ---- kernel 5: readout = mem_v x sparse W, m-sorted sequential gather ----
__global__ __launch_bounds__(256) void k_readout(
    const float* __restrict__ mem_v,
    const int* __restrict__ e_mq, const float* __restrict__ e_w,
    float* __restrict__ out) {
  __shared__ float acc[CVB * Qd];                 // 4*1620*4B = 25.9 KB
  const int tid = threadIdx.x;
  const int o   = blockIdx.x / (CVd / CVB);
  const int cv0 = (blockIdx.x % (CVd / CVB)) * CVB;
  for (int i = tid; i < CVB * Qd; i += 256) acc[i] = 0.f;
  __syncthreads();
  const float* vb = mem_v + (size_t)(o * CVd + cv0) * Md;
  for (int i = tid; i < NNZ; i += 256) {
    const int mq = e_mq[i];
    const float w = e_w[i];
    const int m = mq & 0x7FFF;
    const int q = mq >> 15;
    #pragma unroll
    for (int j = 0; j < CVB; ++j) {
      atomicAdd(&acc[j * Qd + q], w * vb[(size_t)j * Md + m]);  // ds_add_f32
    }
  }
  __syncthreads();
  const size_t ob = (size_t)(o * CVd + cv0) * Qd;
  for (int i = tid; i < CVB * Qd; i += 256) out[ob + i] = acc[i];
}

// --------------------------------------------------------------------------
extern "C" void kernel_launch(void* const* d_in, const int* in_sizes, int n_in,
                              void* d_out, int out_size, void* d_ws, size_t ws_size,
                              hipStream_t stream) {
  (void)in_sizes; (void)n_in; (void)out_size; (void)ws_size;
  const float* qk    = (const float*)d_in[0];    // [1,64,30,54]
  const float* mem_k = (const float*)d_in[1];    // [1,64,32400]
  const float* mem_v = (const float*)d_in[2];    // [3,512,32400]
  float* out = (float*)d_out;                    // [3,1,512,30,54]

  char* ws = (char*)d_ws;
  size_t off = 0;
  float* a_sq   = (float*)(ws + off); off += (size_t)Md * 4;
  int*   count  = (int*)  (ws + off); off += (size_t)Md * 4;
  float* part_v = (float*)(ws + off); off += (size_t)Qd * NSLICE * TK * 4;
  int*   part_i = (int*)  (ws + off); off += (size_t)Qd * NSLICE * TK * 4;
  int*   tk_idx = (int*)  (ws + off); off += (size_t)NNZ * 4;
  float* tk_w   = (float*)(ws + off); off += (size_t)NNZ * 4;
  int*   e_mq   = (int*)  (ws + off); off += (size_t)NNZ * 4;
  float* e_w    = (float*)(ws + off); off += (size_t)NNZ * 4;  // ~2.85 MB total

  k_asq<<<(Md + 255) / 256, 256, 0, stream>>>(mem_k, a_sq, count);
  k_affinity_topk<<<NQT * NSLICE, 256, 0, stream>>>(qk, mem_k, a_sq, part_v, part_i);
  k_finalize<<<(Qd + 31) / 32, 32, 0, stream>>>(part_v, part_i, tk_idx, tk_w, count);
  k_scan<<<1, 1024, 0, stream>>>(count);
  k_scatter<<<(NNZ + 255) / 256, 256, 0, stream>>>(tk_idx, tk_w, count, e_mq, e_w);
  k_readout<<<Od * (CVd / CVB), 256, 0, stream>>>(mem_v, e_mq, e_w, out);
}